// SoftPattern_57604101374173
// MI455X (gfx1250) — compile-verified
//
#include <hip/hip_runtime.h>
#include <math.h>

typedef __attribute__((ext_vector_type(2))) float v2f;
typedef __attribute__((ext_vector_type(8))) float v8f;

#define PAT        7
#define WD         300
#define TLEN       50000
#define NTILE      (TLEN / 16)     // 3125 tiles of 16 tokens
#define TB_STRIDE  16              // floats per token in band table (11 used)
#define NCHUNK     256
#define CHUNK_LEN  196             // 256*196 = 50176 >= 50000
#define SUMM_F     64              // 49 M + 7 b + 7 sM + 1 sb

#define B_STRIDE   302             // LDS panel stride (floats): 8B aligned rows,
                                   // 302 mod 64 = 46 -> 16 cols hit distinct banks

// ---------------------------------------------------------------------------
// Kernel A: gather embeddings + WMMA f32 16x16x4 einsum over the 11 band
// columns of w (zero-padded to 16 in an LDS panel), then
// sigmoid(z - log||w_ij||) -> band table. One wave32 per 16-token tile.
// Inner loop: global_load_b64 (A) + ds_load_b64 (B) + v_wmma, branch-free.
// ---------------------------------------------------------------------------
__global__ __launch_bounds__(256)
void band_logits_wmma(const int* __restrict__ doc,
                      const float* __restrict__ emb,
                      const float* __restrict__ w,
                      float* __restrict__ tband)
{
    __shared__ float Bp[16 * B_STRIDE];   // 19.3 KB of 320 KB WGP LDS

    // --- cooperative fill of the zero-padded B panel (before any early-out) ---
    for (int idx = threadIdx.x; idx < 16 * WD; idx += 256) {
        const int col = idx / WD;
        const int k   = idx - col * WD;
        float v = 0.0f;
        if (col < 6)        v = w[(col * PAT + col) * WD + k];            // (i,i)
        else if (col < 11)  v = w[((col - 6) * PAT + (col - 5)) * WD + k]; // (i,i+1)
        Bp[col * B_STRIDE + k] = v;
    }
    __syncthreads();

    const int lane = threadIdx.x & 31;
    const int wave = threadIdx.x >> 5;
    const int tile = blockIdx.x * 8 + wave;
    if (tile >= NTILE) return;            // whole-wave guard: EXEC stays full

    const int row   = lane & 15;          // A row (token) / B,D column
    const int hi    = lane >> 4;          // 0 or 1 (K half for A and B)
    const int khalf = hi * 2;
    const int col   = row;
    const bool valid = (col < 11);
    const float* bcol = Bp + col * B_STRIDE;

    // log||w_ij|| from the LDS panel (invalid cols give -inf, never stored)
    float nsq = 0.f;
    for (int k = 0; k < WD; k += 2) {
        v2f wv = *(const v2f*)(bcol + k);
        nsq += wv.x * wv.x + wv.y * wv.y;
    }
    const float lognorm = 0.5f * __logf(nsq);

    // Gathered embedding row for this lane's token
    const int t0 = tile * 16;
    const int d  = doc[t0 + row];
    const float* xrow = emb + (long long)d * WD;

    // D = A(16x4) x B(4x16) + C accumulated over K = 300 (75 WMMA steps)
    v8f acc = {};
    #pragma unroll 10
    for (int k0 = 0; k0 < WD; k0 += 4) {
        v2f a = *(const v2f*)(xrow + k0 + khalf);   // A: (M=row, K=k0+khalf+{0,1})
        v2f b = *(const v2f*)(bcol + k0 + khalf);   // B: (K=k0+khalf+{0,1}, N=col)
        acc = __builtin_amdgcn_wmma_f32_16x16x4_f32(false, a, false, b,
                                                    (short)0, acc, false, false);
    }

    // D layout: VGPR r, lane l -> element (M = r + 8*(l>>4), N = l&15)
    if (valid) {
        #pragma unroll
        for (int r = 0; r < 8; ++r) {
            const int t = t0 + hi * 8 + r;
            const float z = acc[r] - lognorm;
            tband[t * TB_STRIDE + col] = 1.0f / (1.0f + __expf(-z));
        }
    }
}

// ---------------------------------------------------------------------------
// Kernel B: parallel chunked affine scan. Each thread composes CHUNK_LEN
// bidiagonal 7x7 transition maps into (M, b) and accumulates the score
// functional (sM, sb) for its chunk.
// ---------------------------------------------------------------------------
__global__ __launch_bounds__(256)
void chunk_scan(const float* __restrict__ tband, float* __restrict__ summ)
{
    const int c = blockIdx.x * blockDim.x + threadIdx.x;
    if (c >= NCHUNK) return;
    const int tstart = c * CHUNK_LEN;
    const int tend   = (tstart + CHUNK_LEN < TLEN) ? (tstart + CHUNK_LEN) : TLEN;

    float M[PAT][PAT], bv[PAT], sM[PAT];
    #pragma unroll
    for (int i = 0; i < PAT; ++i) {
        #pragma unroll
        for (int j = 0; j < PAT; ++j) M[i][j] = (i == j) ? 1.f : 0.f;
        bv[i] = 0.f; sM[i] = 0.f;
    }
    float sb = 0.f;

    for (int t = tstart; t < tend; ++t) {
        const float* tb = tband + t * TB_STRIDE;       // 64B-aligned row
        const float4 v0 = *(const float4*)(tb);
        const float4 v1 = *(const float4*)(tb + 4);
        const float4 v2 = *(const float4*)(tb + 8);
        const float dg[PAT]     = { v0.x, v0.y, v0.z, v0.w, v1.x, v1.y, 0.f };
        const float up[PAT - 1] = { v1.z, v1.w, v2.x, v2.y, v2.z, 0.f };

        // M = M @ T_t   (descending j so column j-1 is still the old value)
        #pragma unroll
        for (int i = 0; i < PAT; ++i) {
            #pragma unroll
            for (int j = PAT - 1; j >= 0; --j) {
                float v = M[i][j] * dg[j];
                if (j > 0) v += M[i][j - 1] * up[j - 1];
                M[i][j] = v;
            }
        }
        // b = b @ T_t + e0
        #pragma unroll
        for (int j = PAT - 1; j >= 0; --j) {
            float v = bv[j] * dg[j];
            if (j > 0) v += bv[j - 1] * up[j - 1];
            bv[j] = v;
        }
        bv[0] += 1.0f;

        // score functional: score_chunk(h_in) = h_in . sM + sb
        #pragma unroll
        for (int i = 0; i < PAT; ++i) sM[i] += M[i][PAT - 1];
        sb += bv[PAT - 1];
    }

    float* o = summ + c * SUMM_F;
    #pragma unroll
    for (int i = 0; i < PAT; ++i)
        #pragma unroll
        for (int j = 0; j < PAT; ++j) o[i * PAT + j] = M[i][j];
    #pragma unroll
    for (int i = 0; i < PAT; ++i) { o[49 + i] = bv[i]; o[56 + i] = sM[i]; }
    o[63] = sb;
}

// ---------------------------------------------------------------------------
// Kernel C: fold the 256 chunk summaries sequentially (tiny).
// ---------------------------------------------------------------------------
__global__ void final_combine(const float* __restrict__ summ, float* __restrict__ out)
{
    if (threadIdx.x != 0 || blockIdx.x != 0) return;
    float h[PAT] = { 1.f, 0.f, 0.f, 0.f, 0.f, 0.f, 0.f };
    float score = 0.f;
    for (int c = 0; c < NCHUNK; ++c) {
        const float* s = summ + c * SUMM_F;
        // score terms use h BEFORE applying this chunk's map
        float ds = s[63];
        #pragma unroll
        for (int i = 0; i < PAT; ++i) ds += h[i] * s[56 + i];
        score += ds;
        float nh[PAT];
        #pragma unroll
        for (int j = 0; j < PAT; ++j) {
            float v = s[49 + j];
            #pragma unroll
            for (int i = 0; i < PAT; ++i) v += h[i] * s[i * PAT + j];
            nh[j] = v;
        }
        #pragma unroll
        for (int j = 0; j < PAT; ++j) h[j] = nh[j];
    }
    out[0] = score;
}

// ---------------------------------------------------------------------------
extern "C" void kernel_launch(void* const* d_in, const int* in_sizes, int n_in,
                              void* d_out, int out_size, void* d_ws, size_t ws_size,
                              hipStream_t stream)
{
    const int*   doc = (const int*)  d_in[0];
    const float* emb = (const float*)d_in[1];
    const float* w   = (const float*)d_in[2];

    float* tband = (float*)d_ws;                            // 50000*16 f32 = 3.2 MB
    float* summ  = (float*)d_ws + (size_t)TLEN * TB_STRIDE; // 256*64 f32 = 64 KB
    float* out   = (float*)d_out;

    band_logits_wmma<<<(NTILE + 7) / 8, 256, 0, stream>>>(doc, emb, w, tband);
    chunk_scan<<<(NCHUNK + 255) / 256, 256, 0, stream>>>(tband, summ);
    final_combine<<<1, 32, 0, stream>>>(summ, out);
}